// Attn_71322226917754
// MI455X (gfx1250) — compile-verified
//
#include <hip/hip_runtime.h>
#include <math.h>

typedef __attribute__((ext_vector_type(2))) float v2f;
typedef __attribute__((ext_vector_type(8))) float v8f;

#define Bn 16
#define En 512
#define Dn 64
#define Hn 128

__device__ __forceinline__ float dev_tanh(float x) {
#if __has_builtin(__builtin_amdgcn_tanhf)
  return __builtin_amdgcn_tanhf(x);
#elif __has_builtin(__builtin_amdgcn_tanh_f32)
  return __builtin_amdgcn_tanh_f32(x);
#else
  return tanhf(x);
#endif
}

// out[M x 128] = X[M x 128] @ W_slice^T (+ bias), W_slice[n][k] = W[n*256 + w_off + k]
// One wave per 16x16 tile, K=128 in steps of 4 via V_WMMA_F32_16X16X4_F32.
__global__ void proj_kernel(const float* __restrict__ X, const float* __restrict__ W,
                            const float* __restrict__ bias, float* __restrict__ out,
                            int w_off) {
  const int wave = threadIdx.x >> 5;
  const int lane = threadIdx.x & 31;
  const int tile = blockIdx.x * 8 + wave;
  const int m0 = (tile >> 3) << 4;        // M tile (Ntiles = 128/16 = 8)
  const int n0 = (tile & 7) << 4;
  const int hf = lane >> 4;               // half-wave: K offset 0 or 2
  const int ln = lane & 15;

  const float* arow = X + (size_t)(m0 + ln) * Hn + 2 * hf;
  const float* bcol = W + (size_t)(n0 + ln) * (2 * Hn) + w_off + 2 * hf;

  v8f c = {};
#pragma unroll 4
  for (int kb = 0; kb < Hn; kb += 4) {
    v2f a = *(const v2f*)(arow + kb);
    v2f bb = *(const v2f*)(bcol + kb);
    c = __builtin_amdgcn_wmma_f32_16x16x4_f32(false, a, false, bb, (short)0, c,
                                              false, false);
  }
  const float bv = bias ? bias[n0 + ln] : 0.f;
  float* orow = out + (size_t)m0 * Hn + n0 + ln;
#pragma unroll
  for (int j = 0; j < 8; ++j) orow[(size_t)(j + 8 * hf) * Hn] = c[j] + bv;
}

// One block (256 thr) per (b,d): energy over e=0..511, masked softmax -> attn row.
__global__ void energy_softmax_kernel(const float* __restrict__ pe,
                                      const float* __restrict__ pdb,
                                      const float* __restrict__ vw,
                                      const float* __restrict__ vb,
                                      const int* __restrict__ elen_p,
                                      const int* __restrict__ dlen_p,
                                      float* __restrict__ attn) {
  const int bd = blockIdx.x;
  const int b = bd >> 6;
  const int d = bd & 63;
  const int t = threadIdx.x;

  __shared__ float s_pd[Hn];
  __shared__ float s_v[Hn];
  __shared__ float s_red[256];

  if (t < Hn) {
    s_pd[t] = pdb[((size_t)b * Dn + d) * Hn + t];  // already includes W_b
    s_v[t] = vw[t];
  }
  __syncthreads();

  const int elen = elen_p[b];
  const bool drow = d < dlen_p[b];
  float* outr = attn + ((size_t)b * Dn + d) * En;

  if (!drow || elen <= 0) {                 // fully masked row -> attn = 0
    outr[t] = 0.f;
    outr[t + 256] = 0.f;
    return;
  }

  const float vbv = vb[0];
  float en[2];
#pragma unroll
  for (int i = 0; i < 2; ++i) {
    const int e = t + i * 256;
    if (e < elen) {
      const float* per = pe + ((size_t)b * En + e) * Hn;
      float acc = 0.f;
#pragma unroll 4
      for (int k = 0; k < Hn; k += 4) {
        float4 p = *(const float4*)(per + k);
        acc += dev_tanh(p.x + s_pd[k + 0]) * s_v[k + 0];
        acc += dev_tanh(p.y + s_pd[k + 1]) * s_v[k + 1];
        acc += dev_tanh(p.z + s_pd[k + 2]) * s_v[k + 2];
        acc += dev_tanh(p.w + s_pd[k + 3]) * s_v[k + 3];
      }
      en[i] = acc + vbv;
    } else {
      en[i] = -INFINITY;
    }
  }

  // block max
  s_red[t] = fmaxf(en[0], en[1]);
  __syncthreads();
  for (int s = 128; s > 0; s >>= 1) {
    if (t < s) s_red[t] = fmaxf(s_red[t], s_red[t + s]);
    __syncthreads();
  }
  const float mx = s_red[0];
  __syncthreads();

  const float ex0 = (en[0] == -INFINITY) ? 0.f : __expf(en[0] - mx);
  const float ex1 = (en[1] == -INFINITY) ? 0.f : __expf(en[1] - mx);

  // block sum
  s_red[t] = ex0 + ex1;
  __syncthreads();
  for (int s = 128; s > 0; s >>= 1) {
    if (t < s) s_red[t] += s_red[t + s];
    __syncthreads();
  }
  const float inv = 1.0f / s_red[0];  // elen>0 => sum >= 1

  outr[t] = ex0 * inv;
  outr[t + 256] = ex1 * inv;
}

// context[b] (64x128) = attn[b] (64x512) @ enc[b] (512x128); one wave per tile.
__global__ void context_kernel(const float* __restrict__ attn,
                               const float* __restrict__ enc,
                               float* __restrict__ out) {
  const int wave = threadIdx.x >> 5;
  const int lane = threadIdx.x & 31;
  const int tile = blockIdx.x * 8 + wave;  // 512 tiles: 16 b * (4 Mtiles * 8 Ntiles)
  const int b = tile >> 5;
  const int r = tile & 31;
  const int m0 = (r >> 3) << 4;
  const int n0 = (r & 7) << 4;
  const int hf = lane >> 4;
  const int ln = lane & 15;

  const float* arow = attn + ((size_t)b * Dn + m0 + ln) * En + 2 * hf;
  const float* bbase = enc + (size_t)b * En * Hn + n0 + ln;

  v8f c = {};
#pragma unroll 4
  for (int kb = 0; kb < En; kb += 4) {
    v2f a = *(const v2f*)(arow + kb);
    v2f bv;
    bv.x = bbase[(size_t)(kb + 2 * hf) * Hn];
    bv.y = bbase[(size_t)(kb + 2 * hf + 1) * Hn];
    c = __builtin_amdgcn_wmma_f32_16x16x4_f32(false, a, false, bv, (short)0, c,
                                              false, false);
  }
  float* orow = out + ((size_t)b * Dn + m0) * Hn + n0 + ln;
#pragma unroll
  for (int j = 0; j < 8; ++j) orow[(size_t)(j + 8 * hf) * Hn] = c[j];
}

extern "C" void kernel_launch(void* const* d_in, const int* in_sizes, int n_in,
                              void* d_out, int out_size, void* d_ws, size_t ws_size,
                              hipStream_t stream) {
  const float* enc = (const float*)d_in[0];   // (B,E,H)
  const float* dec = (const float*)d_in[1];   // (B,D,H)
  const float* W_w = (const float*)d_in[2];   // (H,2H)
  const float* W_b = (const float*)d_in[3];   // (H)
  const float* v_w = (const float*)d_in[4];   // (1,H)
  const float* v_b = (const float*)d_in[5];   // (1)
  const int* elen = (const int*)d_in[6];      // (B)
  const int* dlen = (const int*)d_in[7];      // (B)
  float* out = (float*)d_out;                 // (B,D,H)

  float* ws = (float*)d_ws;
  float* pe = ws;                                   // B*E*H = 1048576 floats
  float* pd = pe + (size_t)Bn * En * Hn;            // B*D*H = 131072 floats
  float* attn = pd + (size_t)Bn * Dn * Hn;          // B*D*E = 524288 floats

  // pe = enc @ We^T            : 512 blocks * 8 waves = 4096 tiles
  proj_kernel<<<dim3((Bn * En / 16) * (Hn / 16) / 8), dim3(256), 0, stream>>>(
      enc, W_w, nullptr, pe, 0);
  // pd = dec @ Wd^T + W_b      : 64 blocks
  proj_kernel<<<dim3((Bn * Dn / 16) * (Hn / 16) / 8), dim3(256), 0, stream>>>(
      dec, W_w, W_b, pd, Hn);
  // energy + masked softmax    : one block per (b,d)
  energy_softmax_kernel<<<dim3(Bn * Dn), dim3(256), 0, stream>>>(
      pe, pd, v_w, v_b, elen, dlen, attn);
  // context = attn @ enc       : 64 blocks
  context_kernel<<<dim3((Bn * Dn / 16) * (Hn / 16) / 8), dim3(256), 0, stream>>>(
      attn, enc, out);
}